// FeatureGen_6356551598644
// MI455X (gfx1250) — compile-verified
//
#include <hip/hip_runtime.h>
#include <hip/hip_bf16.h>

// CDNA5 / gfx1250, wave32. fp32 WMMA (V_WMMA_F32_16X16X4_F32) GEMM pipeline.
//
//  1) vc1 = relu(h_cell @ W1c + b1c)            [16384,128]
//  2) vn1 = relu(h_net  @ W1n + b1n)            [ 8192,128]
//  3) c1  = relu(vc1 @ Rc1w + Rc1b)             [16384,256]
//  4) n1  = relu(vn1 @ Rn1w + Rn1b)             [ 8192,256]
//  5) cat[:, :128] = relu(vc1 + c1 @ Rc2w + Rc2b)
//  6) vn  = relu(vn1 + n1 @ Rn2w + Rn2b)        [ 8192,128]
//  7) cat[:,128:] = g_nc @ vn     <- LDS-tiled kernel: 512MB HBM stream,
//                                    B shared by 128 rows/block via LDS
//  8) out_cell = relu(cat @ W2c + b2c)          [16384,128]
//  9) out_net  = relu(vn  @ W2n + b2n)          [ 8192,128]
// 10) atomic scatter: out_cell[dst] += out_net[src] over 524288 edges

typedef float v2f __attribute__((ext_vector_type(2)));
typedef float v8f __attribute__((ext_vector_type(8)));

#define NTILES 8    // 8 x 16 = 128 output columns per wave

__device__ __forceinline__ v8f wmma4(v2f a, v2f b, v8f c) {
    return __builtin_amdgcn_wmma_f32_16x16x4_f32(
        false, a, false, b, (short)0, c, false, false);
}

// ---------------------------------------------------------------------------
// Generic wave-level GEMM (small K, weights L2-hot).
// Each wave owns a 16-row x 128-col output tile.
// grid.x = M/64 (4 waves/block), grid.y = N/128.
// ---------------------------------------------------------------------------
template <bool RELU, bool HAS_BIAS, bool HAS_SKIP>
__global__ void __launch_bounds__(128)
gemm_wmma_f32(const float* __restrict__ A, int lda,
              const float* __restrict__ W, int ldw,
              const float* __restrict__ bias,
              const float* __restrict__ skip, int ldskip,
              float* __restrict__ out, int ldo,
              int K)
{
    const int lane = threadIdx.x & 31;
    const int wave = threadIdx.x >> 5;
    const int rowBase = (blockIdx.x * 4 + wave) * 16;
    const int g  = lane >> 4;
    const int ln = lane & 15;
    const int colBase = blockIdx.y * (NTILES * 16);

    v8f acc[NTILES];
    v8f zero = {};
#pragma unroll
    for (int j = 0; j < NTILES; ++j) acc[j] = zero;

    const float* Arow = A + (size_t)(rowBase + ln) * lda + (g * 2);
    const float* Wc   = W + colBase + ln;

    for (int k = 0; k < K; k += 4) {
        v2f a = *(const v2f*)(Arow + k);          // global_load_b64
        const float* Wk = Wc + (size_t)(k + g * 2) * ldw;
#pragma unroll
        for (int j = 0; j < NTILES; ++j) {
            v2f b;
            b.x = Wk[j * 16];
            b.y = Wk[j * 16 + ldw];
            acc[j] = wmma4(a, b, acc[j]);
        }
    }

#pragma unroll
    for (int j = 0; j < NTILES; ++j) {
        const int col = colBase + j * 16 + ln;
        float bv = 0.0f;
        if (HAS_BIAS) bv = bias[col];
#pragma unroll
        for (int r = 0; r < 8; ++r) {
            const int row = rowBase + g * 8 + r;
            float v = acc[j][r] + bv;
            if (HAS_SKIP) v += skip[(size_t)row * ldskip + col];
            if (RELU) v = v > 0.0f ? v : 0.0f;
            out[(size_t)row * ldo + col] = v;
        }
    }
}

// ---------------------------------------------------------------------------
// Big streaming GEMM: out[M,128] = A[M,K] @ B[K,128], K large (8192).
// 256 threads = 8 waves/block, each wave one 16-row strip -> 128 rows share
// every B fetch (B L2 traffic == A HBM traffic). B staged per 64-K block into
// LDS, transposed and padded (stride 68) so each fragment is one conflict-free
// ds_load_b64. Register double-buffer overlaps next stage with compute.
// grid.x = M/128.
// ---------------------------------------------------------------------------
#define KB 64
#define LDSS 68   // padded K-stride per column (floats); banks 4*ln+k all distinct

__global__ void __launch_bounds__(256)
gemm_big_wmma_f32(const float* __restrict__ A, int lda,
                  const float* __restrict__ B,        // [K,128] row-major
                  float* __restrict__ out, int ldo,
                  int K)
{
    __shared__ float ldsB[128 * LDSS];   // ~34 KB

    const int tid  = threadIdx.x;
    const int lane = tid & 31;
    const int wave = tid >> 5;
    const int rowBase = (blockIdx.x * 8 + wave) * 16;
    const int g  = lane >> 4;
    const int ln = lane & 15;

    v8f acc[NTILES];
    v8f zero = {};
#pragma unroll
    for (int j = 0; j < NTILES; ++j) acc[j] = zero;

    const float* Arow = A + (size_t)(rowBase + ln) * lda + (g * 2);

    // Staging map: 2048 float4 per 64x128 block, 8 per thread, coalesced.
    float4 stage[8];
#pragma unroll
    for (int i = 0; i < 8; ++i) {
        const int q = i * 256 + tid;
        stage[i] = *(const float4*)(B + (size_t)(q >> 5) * 128 + (q & 31) * 4);
    }

    for (int k0 = 0; k0 < K; k0 += KB) {
        __syncthreads();   // all waves done reading previous LDS tile
#pragma unroll
        for (int i = 0; i < 8; ++i) {
            const int q = i * 256 + tid;
            const int r = q >> 5;
            const int c = (q & 31) * 4;
            ldsB[(c + 0) * LDSS + r] = stage[i].x;   // transpose into LDS
            ldsB[(c + 1) * LDSS + r] = stage[i].y;
            ldsB[(c + 2) * LDSS + r] = stage[i].z;
            ldsB[(c + 3) * LDSS + r] = stage[i].w;
        }
        if (k0 + KB < K) {
            const float* Bn = B + (size_t)(k0 + KB) * 128;
#pragma unroll
            for (int i = 0; i < 8; ++i) {
                const int q = i * 256 + tid;
                stage[i] = *(const float4*)(Bn + (size_t)(q >> 5) * 128 + (q & 31) * 4);
            }
        }
        __syncthreads();   // staged tile visible

        const float* Ak = Arow + k0;
#pragma unroll 4
        for (int kk = 0; kk < KB; kk += 4) {
            __builtin_prefetch(Ak + kk + 2 * KB, 0, 1);   // global_prefetch_b8
            v2f a = *(const v2f*)(Ak + kk);               // global_load_b64
#pragma unroll
            for (int j = 0; j < NTILES; ++j) {
                v2f b = *(const v2f*)(&ldsB[(j * 16 + ln) * LDSS + kk + 2 * g]); // ds_load_b64
                acc[j] = wmma4(a, b, acc[j]);
            }
        }
    }

#pragma unroll
    for (int j = 0; j < NTILES; ++j) {
        const int col = j * 16 + ln;
#pragma unroll
        for (int r = 0; r < 8; ++r) {
            const int row = rowBase + g * 8 + r;
            out[(size_t)row * ldo + col] = acc[j][r];
        }
    }
}

// ---------------------------------------------------------------------------
// Edge scatter: one wave per edge, 4 floats/lane -> global_atomic_add_f32.
// ---------------------------------------------------------------------------
__global__ void __launch_bounds__(256)
scatter_add_kernel(const int* __restrict__ esrc, const int* __restrict__ edst,
                   const float* __restrict__ vn2, float* __restrict__ outc,
                   int nEdge)
{
    const int idx = blockIdx.x * blockDim.x + threadIdx.x;
    const int e = idx >> 5;
    if (e >= nEdge) return;
    const int c = (idx & 31) * 4;
    const int s = esrc[e];
    const int d = edst[e];
    const float4 v = *(const float4*)(vn2 + (size_t)s * 128 + c);
    float* o = outc + (size_t)d * 128 + c;
    atomicAdd(o + 0, v.x);
    atomicAdd(o + 1, v.y);
    atomicAdd(o + 2, v.z);
    atomicAdd(o + 3, v.w);
}

extern "C" void kernel_launch(void* const* d_in, const int* in_sizes, int n_in,
                              void* d_out, int out_size, void* d_ws, size_t ws_size,
                              hipStream_t stream)
{
    (void)in_sizes; (void)n_in; (void)out_size; (void)ws_size;

    const float* h_cell = (const float*)d_in[0];   // [16384,64]
    const float* h_net  = (const float*)d_in[1];   // [8192,32]
    const float* g_nc   = (const float*)d_in[2];   // [16384,8192]
    const int*   esrc   = (const int*)d_in[3];     // [524288]
    const int*   edst   = (const int*)d_in[4];     // [524288]
    const float* W1c  = (const float*)d_in[5];  const float* b1c  = (const float*)d_in[6];
    const float* W1n  = (const float*)d_in[7];  const float* b1n  = (const float*)d_in[8];
    const float* Rc1w = (const float*)d_in[9];  const float* Rc1b = (const float*)d_in[10];
    const float* Rn1w = (const float*)d_in[11]; const float* Rn1b = (const float*)d_in[12];
    const float* Rc2w = (const float*)d_in[13]; const float* Rc2b = (const float*)d_in[14];
    const float* Rn2w = (const float*)d_in[15]; const float* Rn2b = (const float*)d_in[16];
    const float* W2c  = (const float*)d_in[17]; const float* b2c  = (const float*)d_in[18];
    const float* W2n  = (const float*)d_in[19]; const float* b2n  = (const float*)d_in[20];

    // Workspace layout (floats): ~56 MB total
    float* ws  = (float*)d_ws;
    float* vc1 = ws;                       // 16384*128
    float* vn1 = vc1 + 16384 * 128;        //  8192*128
    float* c1  = vn1 + 8192 * 128;         // 16384*256
    float* n1  = c1  + 16384 * 256;        //  8192*256
    float* cat = n1  + 8192 * 256;         // 16384*256 (concat buffer)
    float* vnb = cat + 16384 * 256;        //  8192*128

    float* out_cell = (float*)d_out;               // 16384*128
    float* out_net  = out_cell + 16384 * 128;      //  8192*128

    const dim3 blk(128);

    gemm_wmma_f32<true, true, false><<<dim3(256, 1), blk, 0, stream>>>(
        h_cell, 64, W1c, 128, b1c, nullptr, 0, vc1, 128, 64);
    gemm_wmma_f32<true, true, false><<<dim3(128, 1), blk, 0, stream>>>(
        h_net, 32, W1n, 128, b1n, nullptr, 0, vn1, 128, 32);
    gemm_wmma_f32<true, true, false><<<dim3(256, 2), blk, 0, stream>>>(
        vc1, 128, Rc1w, 256, Rc1b, nullptr, 0, c1, 256, 128);
    gemm_wmma_f32<true, true, false><<<dim3(128, 2), blk, 0, stream>>>(
        vn1, 128, Rn1w, 256, Rn1b, nullptr, 0, n1, 256, 128);
    gemm_wmma_f32<true, true, true><<<dim3(256, 1), blk, 0, stream>>>(
        c1, 256, Rc2w, 128, Rc2b, vc1, 128, cat, 256, 256);
    gemm_wmma_f32<true, true, true><<<dim3(128, 1), blk, 0, stream>>>(
        n1, 256, Rn2w, 128, Rn2b, vn1, 128, vnb, 128, 256);

    // Big one: cat[:,128:] = g_nc @ vnb   (K = 8192)
    gemm_big_wmma_f32<<<dim3(128, 1), dim3(256), 0, stream>>>(
        g_nc, 8192, vnb, cat + 128, 256, 8192);

    gemm_wmma_f32<true, true, false><<<dim3(256, 1), blk, 0, stream>>>(
        cat, 256, W2c, 128, b2c, nullptr, 0, out_cell, 128, 256);
    gemm_wmma_f32<true, true, false><<<dim3(128, 1), blk, 0, stream>>>(
        vnb, 128, W2n, 128, b2n, nullptr, 0, out_net, 128, 128);

    scatter_add_kernel<<<65536, 256, 0, stream>>>(esrc, edst, out_net, out_cell, 524288);
}